// TnoNaive_61916248539625
// MI455X (gfx1250) — compile-verified
//
#include <hip/hip_runtime.h>

typedef float v2f __attribute__((ext_vector_type(2)));
typedef float v8f __attribute__((ext_vector_type(8)));

#define WAVES          8
#define TILE_CH        16
#define ROWS_PER_WAVE  16
#define ROWS_PER_ITER  (WAVES * ROWS_PER_WAVE)   // 128 seq rows per workgroup iteration

// Fixed problem shape (harness: b=8, n=8192, d=512, fp32)
constexpr int N = 8192;
constexpr int D = 512;

// One workgroup = one (batch, 16-channel tile) column. 8 waves cooperate:
// each wave owns a 16x16 (seq x ch) tile per iteration and computes
//   D = L * Y + carry_in      (L = lower-triangular ones, Y = x*t)
// via a chain of four V_WMMA_F32_16X16X4_F32, carry injected through C.
__global__ __launch_bounds__(256) void tno_wmma_scan_kernel(
    const float* __restrict__ x,   // (b, N, D)
    const float* __restrict__ t,   // (2N-2, D), rows < N used
    float* __restrict__ o)         // (b, N, D)
{
    __shared__ float lds_sums[2][WAVES][TILE_CH];   // double-buffered block sums

    const int tid  = threadIdx.x;
    const int wave = tid >> 5;          // wave32
    const int lane = tid & 31;
    const int hi   = lane >> 4;         // 0: lanes 0-15, 1: lanes 16-31
    const int col  = lane & 15;         // channel within tile / M row id

    const int c0 = blockIdx.x * TILE_CH;    // channel tile base
    const int ib = blockIdx.y;              // batch

    // Per-lane base pointers; all in-loop accesses use immediate offsets.
    // Load rows for this lane: (wave*16 + 4k + 2*hi + j), channel c0+col.
    const size_t lane_ch  = (size_t)(c0 + col);
    const size_t ld_row0  = (size_t)(wave * ROWS_PER_WAVE + 2 * hi);
    const size_t st_row0  = (size_t)(wave * ROWS_PER_WAVE + 8 * hi);

    const float* px = x + (size_t)ib * N * D + ld_row0 * D + lane_ch;
    const float* pt = t + ld_row0 * D + lane_ch;
    float*       po = o + (size_t)ib * N * D + st_row0 * D + lane_ch;

    // Constant scan matrix A (16x4 per ksub). A-layout (32-bit A, 16x4):
    //   lanes 0-15:  M=lane,    VGPR0 -> K=4k+0, VGPR1 -> K=4k+1
    //   lanes 16-31: M=lane-16, VGPR0 -> K=4k+2, VGPR1 -> K=4k+3
    v2f A[4];
#pragma unroll
    for (int k = 0; k < 4; ++k) {
        const int k0 = 4 * k + 2 * hi;
        A[k].x = (k0     <= col) ? 1.0f : 0.0f;
        A[k].y = (k0 + 1 <= col) ? 1.0f : 0.0f;
    }

    float carry = 0.0f;   // per-channel running carry (replicated in lane col and col+16)

    constexpr int ITERS = N / ROWS_PER_ITER;   // 64
    for (int it = 0; it < ITERS; ++it) {
        // ---- gather Y = x*t directly in B-operand layout ----
        // B (4x16 f32): lanes 0-15 col=lane rows {4k,4k+1}; lanes 16-31 rows {4k+2,4k+3}
        float y[4][2];
        float psum = 0.0f;
#pragma unroll
        for (int k = 0; k < 4; ++k) {
#pragma unroll
            for (int j = 0; j < 2; ++j) {
                const int roff = (4 * k + j) * D;              // immediate offset
                const float xv = __builtin_nontemporal_load(px + roff); // streamed once
                const float tv = pt[roff];                              // keep t in L2
                const float yv = xv * tv;
                y[k][j] = yv;
                psum += yv;
            }
        }

        // Prefetch next iteration's tile lines (imm offset +128*D fits IOFFSET).
        if (it + 1 < ITERS) {
#pragma unroll
            for (int k = 0; k < 4; ++k) {
                __builtin_prefetch(px + ROWS_PER_ITER * D + (4 * k) * D, 0, 0);
            }
        }

        // ---- per-channel 16-row block sum: this half + other half (lane^16) ----
        const int other = __builtin_amdgcn_ds_bpermute(
            (lane ^ 16) << 2, __builtin_bit_cast(int, psum));
        const float blocksum = psum + __builtin_bit_cast(float, other);

        // ---- cross-wave exclusive scan (double-buffered LDS, one barrier/iter) ----
        const int buf = it & 1;
        if (hi == 0) lds_sums[buf][wave][col] = blocksum;
        __syncthreads();

        float prefix = 0.0f, total = 0.0f;
#pragma unroll
        for (int w = 0; w < WAVES; ++w) {
            const float s = lds_sums[buf][w][col];
            if (w < wave) prefix += s;
            total += s;
        }

        const float cin = carry + prefix;  // carry into this wave's 16-row tile
        carry += total;                    // identical on all waves -> stays consistent

        // ---- D = L*Y + cin via four chained f32 WMMAs (K=16) ----
        v8f acc;
#pragma unroll
        for (int i = 0; i < 8; ++i) acc[i] = cin;   // C layout replicates N per lane

#pragma unroll
        for (int k = 0; k < 4; ++k) {
            v2f B;
            B.x = y[k][0];
            B.y = y[k][1];
            acc = __builtin_amdgcn_wmma_f32_16x16x4_f32(
                /*neg_a=*/false, A[k], /*neg_b=*/false, B,
                /*c_mod=*/(short)0, acc, /*reuse_a=*/false, /*reuse_b=*/false);
        }

        // ---- store D: VGPR vr -> row (st_row0 + vr), immediate offsets vr*D ----
#pragma unroll
        for (int vr = 0; vr < 8; ++vr) {
            __builtin_nontemporal_store(acc[vr], po + vr * D);
        }

        // advance per-lane streams by 128 rows (single 64-bit add each)
        px += (size_t)ROWS_PER_ITER * D;
        pt += (size_t)ROWS_PER_ITER * D;
        po += (size_t)ROWS_PER_ITER * D;
    }
}

extern "C" void kernel_launch(void* const* d_in, const int* in_sizes, int n_in,
                              void* d_out, int out_size, void* d_ws, size_t ws_size,
                              hipStream_t stream) {
    const float* x = (const float*)d_in[0];   // (b, N, D) f32
    const float* t = (const float*)d_in[1];   // (2N-2, D) f32
    float* o = (float*)d_out;

    const int b = in_sizes[0] / (N * D);      // = 8

    dim3 grid(D / TILE_CH, b);                // 32 x 8 = 256 workgroups
    dim3 block(32 * WAVES);                   // 8 waves (wave32)
    tno_wmma_scan_kernel<<<grid, block, 0, stream>>>(x, t, o);
}